// MessagePassingModel_75926431859371
// MI455X (gfx1250) — compile-verified
//
#include <hip/hip_runtime.h>
#include <math.h>

typedef __attribute__((ext_vector_type(2))) float v2f;
typedef __attribute__((ext_vector_type(8))) float v8f;

static constexpr int kB   = 128;
static constexpr int kN   = 29;
static constexpr int kBN  = kB * kN;      // 3712
static constexpr int kF   = 64;
static constexpr int kK   = 64;
static constexpr int kNE  = kN - 1;       // 28 edges per destination node
static constexpr int kXS  = 8 * kF;       // 512 floats per node (2 parity x 4 comps x F)
static constexpr int kPAD = 516;          // LDS row stride (4r+k mod 64 distinct for r<16)

// ---------------------------------------------------------------------------
// One wave computes a 16x64 fp32 GEMM tile: O = A(16x64) * W(64x64) [+ bias]
// using V_WMMA_F32_16X16X4_F32 (4 N-tiles x 16 K-steps).
// A/O in LDS with row stride lda/ldo; W row-major [k*64+g] in global.
// ---------------------------------------------------------------------------
__device__ __forceinline__ void wave_gemm_16x64(const float* __restrict__ Alds, int lda,
                                                const float* __restrict__ Wg,
                                                float* __restrict__ Olds, int ldo,
                                                const float* __restrict__ bias) {
  const int lane = threadIdx.x & 31;
  const int half = lane >> 4;     // 0: K pair {0,1}; 1: K pair {2,3}
  const int rc   = lane & 15;     // A row / B col / D col
  const int koff = half * 2;
#pragma unroll
  for (int nt = 0; nt < 4; ++nt) {
    v8f c = {};
    const int g0 = nt * 16 + rc;
#pragma unroll
    for (int k0 = 0; k0 < 64; k0 += 4) {
      v2f a, b;
      a.x = Alds[rc * lda + k0 + koff];
      a.y = Alds[rc * lda + k0 + koff + 1];
      b.x = Wg[(k0 + koff) * 64 + g0];
      b.y = Wg[(k0 + koff + 1) * 64 + g0];
      c = __builtin_amdgcn_wmma_f32_16x16x4_f32(false, a, false, b, (short)0, c,
                                                false, false);
    }
    const float bb = bias ? bias[g0] : 0.0f;
#pragma unroll
    for (int v = 0; v < 8; ++v) {
      Olds[(v + half * 8) * ldo + g0] = c[v] + bb;
    }
  }
}

// ---------------------------------------------------------------------------
// x init: x[n,0(regular),0(l=0),f] = embed[Z[n]][f], everything else 0
// ---------------------------------------------------------------------------
__global__ void init_x_kernel(float* __restrict__ x, const float* __restrict__ embed,
                              const int* __restrict__ Z) {
  int idx = blockIdx.x * blockDim.x + threadIdx.x;
  if (idx >= kBN * kXS) return;
  int n = idx >> 9, rem = idx & 511;
  float v = 0.0f;
  if (rem < kF) v = embed[Z[n] * kF + rem];
  x[idx] = v;
}

// ---------------------------------------------------------------------------
// Message + aggregate. One block (64 thr = 2 waves) per destination node.
//   rproj[e] = radial(r_e) @ Wb   (WMMA, rows = edges)
//   msg per edge expanded TP (b-pseudo channel == 0), summed over 28 edges.
// Writes y[n]; x untouched (read by other blocks).
// ---------------------------------------------------------------------------
__global__ __launch_bounds__(64) void msg_kernel(const float* __restrict__ x,
                                                 float* __restrict__ y,
                                                 const float* __restrict__ pos,
                                                 const float* __restrict__ Wb,
                                                 const float* __restrict__ wtp,
                                                 int last) {
  __shared__ float rad[32 * 68];
  __shared__ float rpj[32 * 68];
  __shared__ float eu[kNE * 3];    // unit vectors
  __shared__ float el[kNE * 3];    // log(u1), log(1-u1), fcut

  const int n = blockIdx.x;
  const int b = n / kN, i = n - b * kN;
  const int tid = threadIdx.x;

  if (tid < kNE) {
    int jj = tid + (tid >= i ? 1 : 0);
    int src = b * kN + jj;
    float dx = pos[src * 3 + 0] - pos[n * 3 + 0];
    float dy = pos[src * 3 + 1] - pos[n * 3 + 1];
    float dz = pos[src * 3 + 2] - pos[n * 3 + 2];
    float r = sqrtf(dx * dx + dy * dy + dz * dz + 1e-12f);
    float inv = 1.0f / r;
    eu[tid * 3 + 0] = dx * inv;
    eu[tid * 3 + 1] = dy * inv;
    eu[tid * 3 + 2] = dz * inv;
    float u1 = 1.0f / (1.0f + r);
    el[tid * 3 + 0] = logf(u1);
    el[tid * 3 + 1] = logf(fmaxf(1.0f - u1, 1e-12f));
    float x2 = (r * 0.2f) * (r * 0.2f);
    el[tid * 3 + 2] = (x2 < 1.0f) ? expf(1.0f - 1.0f / fmaxf(1.0f - x2, 1e-12f)) : 0.0f;
  }
  __syncthreads();

  {  // radial basis: thread t owns k = t
    float kf = (float)tid;
    float logc = lgammaf(64.0f) - lgammaf(kf + 1.0f) - lgammaf(64.0f - kf);
    for (int j = 0; j < kNE; ++j) {
      float lb = logc + kf * el[j * 3 + 0] + (63.0f - kf) * el[j * 3 + 1];
      rad[j * 68 + tid] = expf(lb) * el[j * 3 + 2];
    }
    for (int j = kNE; j < 32; ++j) rad[j * 68 + tid] = 0.0f;
  }
  __syncthreads();

  {  // rproj = rad(32x64) @ Wb(64x64); wave w handles rows 16w..16w+15
    int w = tid >> 5;
    wave_gemm_16x64(rad + w * 16 * 68, 68, Wb, rpj + w * 16 * 68, 68, nullptr);
  }
  __syncthreads();

  // TP message accumulation: thread t owns feature f = t
  float w[20];
#pragma unroll
  for (int p = 0; p < 20; ++p) w[p] = wtp[p * kF + tid];
  float acc0 = 0, acc1 = 0, acc2 = 0, acc3 = 0, acc4 = 0, acc5 = 0, acc6 = 0, acc7 = 0;
  for (int j = 0; j < kNE; ++j) {
    int jj = j + (j >= i ? 1 : 0);
    const float* xs = x + (size_t)(b * kN + jj) * kXS + tid;
    float sR = xs[0], vRx = xs[64], vRy = xs[128], vRz = xs[192];
    float sP = xs[256], vPx = xs[320], vPy = xs[384], vPz = xs[448];
    float t = rpj[j * 68 + tid];
    float ux = eu[j * 3 + 0], uy = eu[j * 3 + 1], uz = eu[j * 3 + 2];
    float dotR = vRx * ux + vRy * uy + vRz * uz;
    float dotP = vPx * ux + vPy * uy + vPz * uz;
    acc0 += (w[0] * sR + w[3] * dotR) * t;
    acc4 += (w[10] * sP + w[13] * dotP) * t;
    if (!last) {
      float cPx = vPy * uz - vPz * uy, cPy = vPz * ux - vPx * uz, cPz = vPx * uy - vPy * ux;
      float cRx = vRy * uz - vRz * uy, cRy = vRz * ux - vRx * uz, cRz = vRx * uy - vRy * ux;
      acc1 += (w[1] * sR * ux + w[2] * vRx + w[14] * cPx) * t;
      acc2 += (w[1] * sR * uy + w[2] * vRy + w[14] * cPy) * t;
      acc3 += (w[1] * sR * uz + w[2] * vRz + w[14] * cPz) * t;
      acc5 += (w[11] * sP * ux + w[12] * vPx + w[4] * cRx) * t;
      acc6 += (w[11] * sP * uy + w[12] * vPy + w[4] * cRy) * t;
      acc7 += (w[11] * sP * uz + w[12] * vPz + w[4] * cRz) * t;
    }
  }
  float* yo = y + (size_t)n * kXS + tid;
  yo[0] = acc0; yo[64] = acc1; yo[128] = acc2; yo[192] = acc3;
  yo[256] = acc4; yo[320] = acc5; yo[384] = acc6; yo[448] = acc7;
}

// ---------------------------------------------------------------------------
// Per 16-node tile (256 thr = 8 waves, wave = one of 8 components):
// x1=x+y; h=silu(dense1(x1)); h2=dense2(h); x2=h2+y; x = x2 + tp(x2, E-field)
// ---------------------------------------------------------------------------
__global__ __launch_bounds__(256) void dense_kernel(float* __restrict__ x,
                                                    const float* __restrict__ y,
                                                    const float* __restrict__ W1,
                                                    const float* __restrict__ b1,
                                                    const float* __restrict__ W2,
                                                    const float* __restrict__ b2,
                                                    const float* __restrict__ wtf,
                                                    const float* __restrict__ Ef) {
  __shared__ float bufA[16 * kPAD];
  __shared__ float bufB[16 * kPAD];
  const int tid = threadIdx.x;
  const int n0 = blockIdx.x * 16;

  for (int idx = tid; idx < 16 * kXS; idx += 256) {
    int node = idx >> 9, rem = idx & 511;
    size_t g = (size_t)(n0 + node) * kXS + rem;
    bufA[node * kPAD + rem] = x[g] + y[g];
  }
  __syncthreads();
  {  // dense1
    int c = tid >> 5, p = c >> 2, l = (c & 3) ? 1 : 0;
    wave_gemm_16x64(bufA + c * 64, kPAD, W1 + (p * 2 + l) * 4096, bufB + c * 64, kPAD,
                    (l == 0) ? (b1 + p * 64) : nullptr);
  }
  __syncthreads();
  // silu: gate by same-parity scalar channel (read bufB, write bufA)
  for (int idx = tid; idx < 16 * kXS; idx += 256) {
    int node = idx >> 9, rem = idx & 511;
    int c = rem >> 6, f = rem & 63, p = c >> 2;
    float s = bufB[node * kPAD + (p * 4) * 64 + f];
    float g = 1.0f / (1.0f + expf(-s));
    bufA[node * kPAD + rem] = bufB[node * kPAD + rem] * g;
  }
  __syncthreads();
  {  // dense2
    int c = tid >> 5, p = c >> 2, l = (c & 3) ? 1 : 0;
    wave_gemm_16x64(bufA + c * 64, kPAD, W2 + (p * 2 + l) * 4096, bufB + c * 64, kPAD,
                    (l == 0) ? (b2 + p * 64) : nullptr);
  }
  __syncthreads();
  // x2 = h2 + y
  for (int idx = tid; idx < 16 * kXS; idx += 256) {
    int node = idx >> 9, rem = idx & 511;
    bufA[node * kPAD + rem] =
        bufB[node * kPAD + rem] + y[(size_t)(n0 + node) * kXS + rem];
  }
  __syncthreads();
  // field TP: x = x2 + tp(x2, xEF)  (xEF: s2=1, v2=E on both parity channels)
  for (int idx = tid; idx < 16 * 64; idx += 256) {
    int node = idx >> 6, f = idx & 63;
    int gn = n0 + node, bb = gn / kN;
    float Ex = Ef[bb * 3 + 0], Ey = Ef[bb * 3 + 1], Ez = Ef[bb * 3 + 2];
    const float* a = bufA + node * kPAD + f;
    float sR = a[0], vRx = a[64], vRy = a[128], vRz = a[192];
    float sP = a[256], vPx = a[320], vPy = a[384], vPz = a[448];
    float w[20];
#pragma unroll
    for (int p = 0; p < 20; ++p) w[p] = wtf[p * kF + f];
    float dR = vRx * Ex + vRy * Ey + vRz * Ez;
    float dP = vPx * Ex + vPy * Ey + vPz * Ez;
    float cRx = vRy * Ez - vRz * Ey, cRy = vRz * Ex - vRx * Ez, cRz = vRx * Ey - vRy * Ex;
    float cPx = vPy * Ez - vPz * Ey, cPy = vPz * Ex - vPx * Ez, cPz = vPx * Ey - vPy * Ex;
    float* xo = x + (size_t)gn * kXS + f;
    xo[0]   = sR + w[0] * sR + w[3] * dR + w[15] * sP + w[18] * dP;
    xo[64]  = vRx + w[1] * sR * Ex + w[2] * vRx + w[16] * sP * Ex + w[17] * vPx + w[9] * cRx + w[14] * cPx;
    xo[128] = vRy + w[1] * sR * Ey + w[2] * vRy + w[16] * sP * Ey + w[17] * vPy + w[9] * cRy + w[14] * cPy;
    xo[192] = vRz + w[1] * sR * Ez + w[2] * vRz + w[16] * sP * Ez + w[17] * vPz + w[9] * cRz + w[14] * cPz;
    xo[256] = sP + w[5] * sR + w[8] * dR + w[10] * sP + w[13] * dP;
    xo[320] = vPx + w[6] * sR * Ex + w[7] * vRx + w[11] * sP * Ex + w[12] * vPx + w[4] * cRx + w[19] * cPx;
    xo[384] = vPy + w[6] * sR * Ey + w[7] * vRy + w[11] * sP * Ey + w[12] * vPy + w[4] * cRy + w[19] * cPy;
    xo[448] = vPz + w[6] * sR * Ez + w[7] * vRz + w[11] * sP * Ez + w[12] * vPz + w[4] * cRz + w[19] * cPz;
  }
}

// ---------------------------------------------------------------------------
// x = tp(x, dense_td(x), td_tp_w): WMMA dense then full 20-path TP.
// ---------------------------------------------------------------------------
__global__ __launch_bounds__(256) void td_kernel(float* __restrict__ x,
                                                 const float* __restrict__ Wtd,
                                                 const float* __restrict__ wtp) {
  __shared__ float bufA[16 * kPAD];
  __shared__ float bufB[16 * kPAD];
  const int tid = threadIdx.x;
  const int n0 = blockIdx.x * 16;

  for (int idx = tid; idx < 16 * kXS; idx += 256) {
    int node = idx >> 9, rem = idx & 511;
    bufA[node * kPAD + rem] = x[(size_t)(n0 + node) * kXS + rem];
  }
  __syncthreads();
  {
    int c = tid >> 5, p = c >> 2, l = (c & 3) ? 1 : 0;
    wave_gemm_16x64(bufA + c * 64, kPAD, Wtd + (p * 2 + l) * 4096, bufB + c * 64, kPAD,
                    nullptr);
  }
  __syncthreads();
  for (int idx = tid; idx < 16 * 64; idx += 256) {
    int node = idx >> 6, f = idx & 63;
    const float* a = bufA + node * kPAD + f;
    const float* bptr = bufB + node * kPAD + f;
    float sR = a[0], vRx = a[64], vRy = a[128], vRz = a[192];
    float sP = a[256], vPx = a[320], vPy = a[384], vPz = a[448];
    float tR = bptr[0], uRx = bptr[64], uRy = bptr[128], uRz = bptr[192];
    float tP = bptr[256], uPx = bptr[320], uPy = bptr[384], uPz = bptr[448];
    float w[20];
#pragma unroll
    for (int p = 0; p < 20; ++p) w[p] = wtp[p * kF + f];
    float dRR = vRx * uRx + vRy * uRy + vRz * uRz;
    float dRP = vRx * uPx + vRy * uPy + vRz * uPz;
    float dPR = vPx * uRx + vPy * uRy + vPz * uRz;
    float dPP = vPx * uPx + vPy * uPy + vPz * uPz;
    float cRRx = vRy * uRz - vRz * uRy, cRRy = vRz * uRx - vRx * uRz, cRRz = vRx * uRy - vRy * uRx;
    float cRPx = vRy * uPz - vRz * uPy, cRPy = vRz * uPx - vRx * uPz, cRPz = vRx * uPy - vRy * uPx;
    float cPRx = vPy * uRz - vPz * uRy, cPRy = vPz * uRx - vPx * uRz, cPRz = vPx * uRy - vPy * uRx;
    float cPPx = vPy * uPz - vPz * uPy, cPPy = vPz * uPx - vPx * uPz, cPPz = vPx * uPy - vPy * uPx;
    float* xo = x + (size_t)(n0 + node) * kXS + f;
    xo[0]   = w[0] * sR * tR + w[3] * dRR + w[15] * sP * tP + w[18] * dPP;
    xo[64]  = w[1] * sR * uRx + w[2] * vRx * tR + w[16] * sP * uPx + w[17] * vPx * tP + w[9] * cRPx + w[14] * cPRx;
    xo[128] = w[1] * sR * uRy + w[2] * vRy * tR + w[16] * sP * uPy + w[17] * vPy * tP + w[9] * cRPy + w[14] * cPRy;
    xo[192] = w[1] * sR * uRz + w[2] * vRz * tR + w[16] * sP * uPz + w[17] * vPz * tP + w[9] * cRPz + w[14] * cPRz;
    xo[256] = w[5] * sR * tP + w[8] * dRP + w[10] * sP * tR + w[13] * dPR;
    xo[320] = w[6] * sR * uPx + w[7] * vRx * tP + w[11] * sP * uRx + w[12] * vPx * tR + w[4] * cRRx + w[19] * cPPx;
    xo[384] = w[6] * sR * uPy + w[7] * vRy * tP + w[11] * sP * uRy + w[12] * vPy * tR + w[4] * cRRy + w[19] * cPPy;
    xo[448] = w[6] * sR * uPz + w[7] * vRz * tP + w[11] * sP * uRz + w[12] * vPz * tR + w[4] * cRRz + w[19] * cPPz;
  }
}

// ---------------------------------------------------------------------------
// energy[b] = sum_i ( x[b,i,0,0,:] . out_w + element_bias[Z[b,i]] )
// ---------------------------------------------------------------------------
__global__ __launch_bounds__(64) void out_kernel(const float* __restrict__ x,
                                                 const float* __restrict__ ow,
                                                 const float* __restrict__ eb,
                                                 const int* __restrict__ Z,
                                                 float* __restrict__ energy) {
  __shared__ float red[64];
  const int b = blockIdx.x, t = threadIdx.x;
  float acc = 0.0f;
  for (int i = 0; i < kN; ++i) acc += x[(size_t)(b * kN + i) * kXS + t];
  float val = acc * ow[t];
  if (t < kN) val += eb[Z[b * kN + t]];
  red[t] = val;
  __syncthreads();
  for (int s = 32; s > 0; s >>= 1) {
    if (t < s) red[t] += red[t + s];
    __syncthreads();
  }
  if (t == 0) energy[b] = red[0];
}

extern "C" void kernel_launch(void* const* d_in, const int* in_sizes, int n_in,
                              void* d_out, int out_size, void* d_ws, size_t ws_size,
                              hipStream_t stream) {
  (void)in_sizes; (void)n_in; (void)out_size; (void)ws_size;
  const int*   Z     = (const int*)d_in[0];
  const float* pos   = (const float*)d_in[1];
  const float* Ef    = (const float*)d_in[2];
  // d_in[3]/d_in[4]: dst_idx/src_idx — graph is meshgrid(ii != jj); handled structurally
  const float* embed = (const float*)d_in[5];
  const float* mpb   = (const float*)d_in[6];
  const float* mpt   = (const float*)d_in[7];
  const float* d1w   = (const float*)d_in[8];
  const float* d1b   = (const float*)d_in[9];
  const float* d2w   = (const float*)d_in[10];
  const float* d2b   = (const float*)d_in[11];
  const float* tw    = (const float*)d_in[12];
  const float* tdw   = (const float*)d_in[13];
  const float* tdt   = (const float*)d_in[14];
  const float* ow    = (const float*)d_in[15];
  const float* eb    = (const float*)d_in[16];
  float* energy = (float*)d_out;

  float* xb = (float*)d_ws;
  float* yb = xb + (size_t)kBN * kXS;

  init_x_kernel<<<(kBN * kXS + 255) / 256, 256, 0, stream>>>(xb, embed, Z);
  for (int i = 0; i < 2; ++i) {
    msg_kernel<<<kBN, 64, 0, stream>>>(xb, yb, pos, mpb + i * 4096, mpt + i * 1280,
                                       (i == 1) ? 1 : 0);
    dense_kernel<<<kBN / 16, 256, 0, stream>>>(xb, yb, d1w + i * 16384, d1b + i * 128,
                                               d2w + i * 16384, d2b + i * 128,
                                               tw + i * 1280, Ef);
    td_kernel<<<kBN / 16, 256, 0, stream>>>(xb, tdw + i * 16384, tdt + i * 1280);
  }
  out_kernel<<<kB, 64, 0, stream>>>(xb, ow, eb, Z, energy);
}